// SparseAttention_11665131176600
// MI455X (gfx1250) — compile-verified
//
#include <hip/hip_runtime.h>

typedef float v2f __attribute__((ext_vector_type(2)));
typedef float v8f __attribute__((ext_vector_type(8)));

#define BLK   32
#define DIM   64
#define NB    512     // S / BLK = 16384 / 32
#define BH    4       // B * H
#define SPAT  16384   // 16*32*32

// One wave32 per (query block row, batch-head).
// Phase 1: S = (Q K^T) * scale  -> masked -> LDS  (v_wmma_f32_16x16x4_f32)
// Phase 2: row softmax in LDS (lane = query row)
// Phase 3: O = P V                              (v_wmma_f32_16x16x4_f32)
__global__ __launch_bounds__(32)
void sparse_attn_f32_wmma(const float* __restrict__ q,
                          const float* __restrict__ k,
                          const float* __restrict__ v,
                          const int* __restrict__ col_idx,
                          const unsigned char* __restrict__ valid,
                          float* __restrict__ out,
                          int K, int rs /* LDS row stride in floats */) {
    extern __shared__ float smem[];              // [32][rs]
    const int row  = blockIdx.x;                 // query block index (0..NB-1)
    const int bh   = blockIdx.y;                 // batch-head
    const int lane = threadIdx.x;                // 0..31

    const long base = (long)bh * SPAT * DIM;
    const float* Q  = q + base + (long)row * BLK * DIM;
    const float* Kp = k + base;
    const float* Vp = v + base;

    // valid[] is prefix-true; count active key blocks (uniform across the wave)
    int nv = 0;
    for (int j = 0; j < K; ++j) nv += (valid[row * K + j] != 0) ? 1 : 0;

    const int lrow  = lane & 15;                 // M (or N) index inside a 16-tile
    const int khalf = (lane >> 4) << 1;          // K offset 0 or 2 (A/B frag layout)
    const int mhalf = (lane >> 4) << 3;          // +8 rows for C/D frag upper half

    // ---- preload Q as WMMA A-fragments: qa[mtile][kchunk] ----
    v2f qa[2][16];
#pragma unroll
    for (int t = 0; t < 2; ++t)
#pragma unroll
        for (int c = 0; c < 16; ++c) {
            const float* p = Q + (t * 16 + lrow) * DIM + c * 4 + khalf;
            v2f a; a[0] = p[0]; a[1] = p[1];
            qa[t][c] = a;
        }

    const float scale = 0.125f;                  // 1/sqrt(64)

    // ================= Phase 1: scores -> LDS =================
    for (int j = 0; j < nv; ++j) {
        const int  c    = col_idx[row * K + j];
        const bool diag = (c == row);
        const float* Kb = Kp + (long)c * BLK * DIM;
#pragma unroll
        for (int u = 0; u < 2; ++u) {            // key 16-tile within block
            v2f kf[16];                          // B-frags of K^T == row loads of K
#pragma unroll
            for (int cc = 0; cc < 16; ++cc) {
                const float* p = Kb + (u * 16 + lrow) * DIM + cc * 4 + khalf;
                v2f b; b[0] = p[0]; b[1] = p[1];
                kf[cc] = b;
            }
#pragma unroll
            for (int t = 0; t < 2; ++t) {        // query 16-tile
                v8f acc = {0.f, 0.f, 0.f, 0.f, 0.f, 0.f, 0.f, 0.f};
#pragma unroll
                for (int cc = 0; cc < 16; ++cc)
                    acc = __builtin_amdgcn_wmma_f32_16x16x4_f32(
                        false, qa[t][cc], false, kf[cc], (short)0, acc, false, false);
                const int ncol = u * 16 + lrow;  // key index within block
#pragma unroll
                for (int g = 0; g < 8; ++g) {
                    const int m = t * 16 + g + mhalf;        // query index
                    float s = acc[g] * scale;
                    if (diag && (ncol > m)) s = -1e30f;      // causal mask on diag blk
                    smem[m * rs + j * BLK + ncol] = s;
                }
            }
        }
    }
    __syncthreads();

    // ================= Phase 2: softmax (lane = query row) =================
    {
        const int ncols = nv * BLK;
        float* rowp = smem + lane * rs;          // stride 641 floats -> no bank conflict
        float mx = -3.0e38f;
        for (int cidx = 0; cidx < ncols; ++cidx) mx = fmaxf(mx, rowp[cidx]);
        float sum = 0.f;
        for (int cidx = 0; cidx < ncols; ++cidx) {
            float e = __expf(rowp[cidx] - mx);
            rowp[cidx] = e;
            sum += e;
        }
        const float inv = 1.0f / sum;
        for (int cidx = 0; cidx < ncols; ++cidx) rowp[cidx] *= inv;
    }
    __syncthreads();

    // ================= Phase 3: O = P * V =================
    v8f o[2][4];
#pragma unroll
    for (int t = 0; t < 2; ++t)
#pragma unroll
        for (int u = 0; u < 4; ++u)
            o[t][u] = (v8f){0.f, 0.f, 0.f, 0.f, 0.f, 0.f, 0.f, 0.f};

    for (int j = 0; j < nv; ++j) {
        const int  c    = col_idx[row * K + j];
        const float* Vb = Vp + (long)c * BLK * DIM;
#pragma unroll
        for (int c2 = 0; c2 < 8; ++c2) {         // K2 = 32 keys -> 8 chunks of 4
            v2f pa[2];
#pragma unroll
            for (int t = 0; t < 2; ++t) {        // P A-frags from LDS
                const float* p = smem + (t * 16 + lrow) * rs + j * BLK + c2 * 4 + khalf;
                v2f a; a[0] = p[0]; a[1] = p[1];
                pa[t] = a;
            }
#pragma unroll
            for (int u = 0; u < 4; ++u) {        // output 16-tile in D
                const int kk   = c2 * 4 + khalf;
                const float* p = Vb + kk * DIM + u * 16 + lrow;
                v2f b; b[0] = p[0]; b[1] = p[DIM];   // B-frag of V (stride-D column pair)
#pragma unroll
                for (int t = 0; t < 2; ++t)
                    o[t][u] = __builtin_amdgcn_wmma_f32_16x16x4_f32(
                        false, pa[t], false, b, (short)0, o[t][u], false, false);
            }
        }
    }

    // ---- store O (C/D fragment layout) ----
    float* O = out + base + (long)row * BLK * DIM;
#pragma unroll
    for (int t = 0; t < 2; ++t)
#pragma unroll
        for (int u = 0; u < 4; ++u)
#pragma unroll
            for (int g = 0; g < 8; ++g) {
                const int m = t * 16 + g + mhalf;
                O[m * DIM + u * 16 + lrow] = o[t][u][g];
            }
}

extern "C" void kernel_launch(void* const* d_in, const int* in_sizes, int n_in,
                              void* d_out, int out_size, void* d_ws, size_t ws_size,
                              hipStream_t stream) {
    const float*         q       = (const float*)d_in[0];
    const float*         k       = (const float*)d_in[1];
    const float*         v       = (const float*)d_in[2];
    const int*           col_idx = (const int*)d_in[3];
    const unsigned char* valid   = (const unsigned char*)d_in[4]; // jax bool = 1 byte
    float*               out     = (float*)d_out;

    const int K  = in_sizes[3] / NB;        // padded active-block count (20 here)
    const int rs = K * BLK + 1;             // odd row stride -> bank-conflict free
    const size_t smem = (size_t)BLK * rs * sizeof(float);   // ~80 KB, <= 320 KB/WGP

    dim3 grid(NB, BH);
    sparse_attn_f32_wmma<<<grid, dim3(32), smem, stream>>>(
        q, k, v, col_idx, valid, out, K, rs);
}